// Conv2DCollapse_w_pillar_6614249636315
// MI455X (gfx1250) — compile-verified
//
#include <hip/hip_runtime.h>
#include <stdint.h>

// Problem constants (fixed by the reference's setup_inputs()).
constexpr int B  = 8;
constexpr int C  = 64;
constexpr int NX = 512;
constexpr int NY = 512;
constexpr int CELLS = B * NY * NX;          // 2,097,152 cells
// Workspace layout: d_ws[0 .. CELLS) = int32 occupancy map (pillar id or -1), 8 MB.

typedef __attribute__((address_space(3))) int   lds_int;
typedef __attribute__((address_space(3))) float lds_float;

typedef unsigned int v4u __attribute__((ext_vector_type(4)));
typedef int          v8i __attribute__((ext_vector_type(8)));
typedef int          v4i __attribute__((ext_vector_type(4)));

// ---------------------------------------------------------------------------
// Pass 1: init occupancy map to -1 (8 MB, int4 stores).
// ---------------------------------------------------------------------------
__global__ void init_map_kernel(int* __restrict__ map) {
    int i = blockIdx.x * blockDim.x + threadIdx.x;     // one int4 per thread
    if (i * 4 < CELLS) {
        reinterpret_cast<int4*>(map)[i] = make_int4(-1, -1, -1, -1);
    }
}

// ---------------------------------------------------------------------------
// Pass 2: scatter pillar ids into the map (tiny: 200k 4B stores).
// ---------------------------------------------------------------------------
__global__ void scatter_ids_kernel(const int* __restrict__ coords,
                                   int* __restrict__ map, int N) {
    int n = blockIdx.x * blockDim.x + threadIdx.x;
    if (n >= N) return;
    int4 c = reinterpret_cast<const int4*>(coords)[n];          // [b, z, y, x]
    long long cell = (long long)c.x * (NX * NY)                 // b*(nx*ny)
                   + (long long)c.y                              // + z
                   + (long long)c.z * NX                         // + y*nx
                   + (long long)c.w;                             // + x
    if (cell >= 0 && cell < CELLS) map[(size_t)cell] = n;
}

// ---------------------------------------------------------------------------
// Pass 3: build (64c x 256x) tiles in LDS, stream them out with the TDM.
//   - map chunk staged via gfx1250 async global->LDS DMA
//   - pillar rows read with contiguous b128 loads, transposed into LDS
//     tile[c][x] (lane-consecutive writes: bank-conflict free)
//   - each of the 8 waves issues one TENSOR_STORE_FROM_LDS for its 8
//     c-planes: tile_dim=(256,8), tensor_dim0_stride=NY*NX elements,
//     then waits TENSORcnt==0 before retiring (LDS lifetime).
// ---------------------------------------------------------------------------
__global__ void gather_bev_kernel(const float* __restrict__ feat,
                                  const int* __restrict__ map,
                                  float* __restrict__ out) {
    const int tid = threadIdx.x;                 // 0..255
    const int x0  = blockIdx.x * 256;            // 0 or 256
    const int x   = x0 + tid;
    const int y   = blockIdx.y;                  // 0..511
    const int b   = blockIdx.z;                  // 0..7

    __shared__ int   smap[256];
    __shared__ float tile[C * 256];              // [c][x], 64 KB

    // --- async global->LDS staging of this block's map row chunk ---
    const int* gsrc = map + (((size_t)b * NY + y) * NX + x);
    unsigned lds_off = (unsigned)(unsigned long long)(lds_int*)(&smap[tid]);
    unsigned long long gaddr_map = (unsigned long long)(uintptr_t)gsrc;
    asm volatile("global_load_async_to_lds_b32 %0, %1, off"
                 :: "v"(lds_off), "v"(gaddr_map)
                 : "memory");
    asm volatile("s_wait_asynccnt 0" ::: "memory");
    __syncthreads();

    const int n = smap[tid];

    if (n >= 0) {
        const float* frow = feat + (size_t)n * C;        // contiguous 256 B
        __builtin_prefetch(frow, 0, 1);                  // global_prefetch_b8
        __builtin_prefetch(frow + 32, 0, 1);
#pragma unroll
        for (int c = 0; c < C; c += 4) {
            float4 f = *reinterpret_cast<const float4*>(frow + c);
            tile[(c + 0) * 256 + tid] = f.x;
            tile[(c + 1) * 256 + tid] = f.y;
            tile[(c + 2) * 256 + tid] = f.z;
            tile[(c + 3) * 256 + tid] = f.w;
        }
    } else {
#pragma unroll
        for (int c = 0; c < C; ++c) tile[c * 256 + tid] = 0.0f;
    }
    __syncthreads();

    // --- per-wave TDM store: wave w owns c-planes [8w, 8w+8) ---
    const int wave = __builtin_amdgcn_readfirstlane(tid >> 5);   // 0..7, SGPR

#if __has_builtin(__builtin_amdgcn_tensor_store_from_lds)
    const unsigned lds_base =
        (unsigned)(unsigned long long)(lds_float*)(&tile[0]);
    const unsigned lds_addr = lds_base + (unsigned)wave * (8u * 256u * 4u);

    const unsigned long long gaddr = (unsigned long long)(uintptr_t)
        (out + ((((size_t)b * C) + (size_t)wave * 8) * NY + (size_t)y) * NX + x0);

    // ---- D# group 0 (128b): count=1 | lds_addr | global_addr(57b) | type=2
    v4u g0;
    g0[0] = 1u;                                          // count = 1 (valid)
    g0[1] = lds_addr;                                    // lds_addr (bytes)
    g0[2] = (unsigned)(gaddr & 0xFFFFFFFFull);           // global_addr[31:0]
    g0[3] = (unsigned)((gaddr >> 32) & 0x1FFFFFFull)     // global_addr[56:32]
          | (2u << 30);                                  // type = 2 ("image")

    // ---- D# group 1 (256b): data_size=4B, 2-D tile 256x8, stride NY*NX
    const unsigned td0 = 256u;                           // tensor_dim0 (x)
    const unsigned td1 = 8u;                             // tensor_dim1 (c rows)
    const unsigned tile0 = 256u, tile1 = 8u, tile2 = 0u;
    const unsigned stride0 = (unsigned)(NY * NX);        // 262144 elements
    v8i g1;
    g1[0] = (int)(2u << 16);                             // data_size = 2 -> 4B
    g1[1] = (int)((td0 & 0xFFFFu) << 16);                // tensor_dim0[15:0]
    g1[2] = (int)(((td0 >> 16) & 0xFFFFu)                // tensor_dim0[31:16]
          |       ((td1 & 0xFFFFu) << 16));              // tensor_dim1[15:0]
    g1[3] = (int)(((td1 >> 16) & 0xFFFFu)                // tensor_dim1[31:16]
          |       ((tile0 & 0xFFFFu) << 16));            // tile_dim0
    g1[4] = (int)((tile1 & 0xFFFFu)                      // tile_dim1
          |       ((tile2 & 0xFFFFu) << 16));            // tile_dim2
    g1[5] = (int)stride0;                                // dim0_stride[31:0]
    g1[6] = 0;                                           // dim0_stride[47:32]
    g1[7] = 0;                                           // dim1_stride (2-D)

    v4i gz4;
    gz4[0] = 0; gz4[1] = 0; gz4[2] = 0; gz4[3] = 0;      // groups 2/3: 2-D

    v8i gz8;
    gz8[0] = 0; gz8[1] = 0; gz8[2] = 0; gz8[3] = 0;      // extra group (6-arg
    gz8[4] = 0; gz8[5] = 0; gz8[6] = 0; gz8[7] = 0;      // clang-23 form)

    __builtin_amdgcn_tensor_store_from_lds(g0, g1, gz4, gz4, gz8, 0);
    __builtin_amdgcn_s_wait_tensorcnt(0);                // LDS lifetime
#else
    // Fallback: direct non-temporal coalesced sweep from LDS.
    const size_t cstride = (size_t)NY * NX;
    size_t obase = (((size_t)b * C) * NY + y) * NX + x;
#pragma unroll 8
    for (int c = 0; c < C; ++c) {
        __builtin_nontemporal_store(tile[c * 256 + tid],
                                    out + obase + (size_t)c * cstride);
    }
    (void)wave;
#endif
}

// ---------------------------------------------------------------------------
extern "C" void kernel_launch(void* const* d_in, const int* in_sizes, int n_in,
                              void* d_out, int out_size, void* d_ws, size_t ws_size,
                              hipStream_t stream) {
    const float* feat   = (const float*)d_in[0];   // (N, 64) f32
    const int*   coords = (const int*)d_in[1];     // (N, 4)  i32 [b,z,y,x]
    const int N = in_sizes[1] / 4;

    int*   map = (int*)d_ws;                       // needs CELLS*4 = 8 MB
    float* out = (float*)d_out;                    // (8, 64, 512, 512) f32

    // Pass 1: map = -1
    init_map_kernel<<<(CELLS / 4 + 255) / 256, 256, 0, stream>>>(map);

    // Pass 2: map[cell] = pillar id
    scatter_ids_kernel<<<(N + 255) / 256, 256, 0, stream>>>(coords, map, N);

    // Pass 3: LDS tile build + TDM streaming store over the full output
    dim3 grid(NX / 256, NY, B);
    gather_bev_kernel<<<grid, 256, 0, stream>>>(feat, map, out);
}